// BatchedResGatedGraphConv_86225763435212
// MI455X (gfx1250) — compile-verified
//
#include <hip/hip_runtime.h>

// Problem constants (match reference)
#define BT   48      // B*T
#define NN   2048    // nodes
#define FF   64      // features in/out
#define EE   32768   // edges
#define LDSS 76      // LDS row stride (floats): conflict-free for A and B frag patterns

typedef __attribute__((ext_vector_type(2))) float v2f;
typedef __attribute__((ext_vector_type(8))) float v8f;
typedef __attribute__((ext_vector_type(4))) unsigned int u32x4;
typedef __attribute__((ext_vector_type(4))) int i32x4;
typedef __attribute__((ext_vector_type(8))) int i32x8;

#if defined(__has_builtin)
#if __has_builtin(__builtin_amdgcn_tensor_load_to_lds) && __has_builtin(__builtin_amdgcn_s_wait_tensorcnt)
#define USE_TDM 1
#endif
#endif
#ifndef USE_TDM
#define USE_TDM 0
#endif

// ---------------------------------------------------------------------------
// Kernel 1: fused projection GEMM via fp32 WMMA.
//   grid = (BT * N/128, 4), block = 256 (8 waves)
//   blockIdx.y selects which weight matrix (0:Wk 1:Wq 2:Wv 3:Wskip)
//   X tile (128x64) is DMA'd into LDS by the Tensor Data Mover with hardware
//   row padding (64 DW rows + 12 DW pad == stride 76). W is staged transposed
//   so both A and B fragments are single ds_load_b64s.
// ---------------------------------------------------------------------------
__global__ __launch_bounds__(256)
void proj_wmma_kernel(const float* __restrict__ x,
                      const float* __restrict__ Wk, const float* __restrict__ bk,
                      const float* __restrict__ Wq, const float* __restrict__ bq,
                      const float* __restrict__ Wv, const float* __restrict__ bv,
                      const float* __restrict__ Ws, const float* __restrict__ bs,
                      float* __restrict__ K, float* __restrict__ Q,
                      float* __restrict__ V, float* __restrict__ Out)
{
    __shared__ float sX[128][LDSS];     // X tile, rows padded to 76 floats
    __shared__ float sWt[64][LDSS];     // sWt[n][k] = W[k][n]  (transposed)

    const int mat  = blockIdx.y;
    const int bt   = blockIdx.x >> 4;         // / (N/128)
    const int rblk = blockIdx.x & 15;
    const int row0 = rblk * 128;

    const float* Wm  = (mat == 0) ? Wk : (mat == 1) ? Wq : (mat == 2) ? Wv : Ws;
    const float* bia = (mat == 0) ? bk : (mat == 1) ? bq : (mat == 2) ? bv : bs;
    float*       dst = (mat == 0) ? K  : (mat == 1) ? Q  : (mat == 2) ? V  : Out;

    const float* xg = x + (((size_t)bt * NN) + row0) * FF;

#if USE_TDM
    // --- Tensor Data Mover: DMA the 128x64 fp32 X tile into LDS, with HW
    //     padding of 12 DWORDs after every 64-DWORD row (-> stride 76).
    if (threadIdx.x == 0) {
        const unsigned long long ga = (unsigned long long)(uintptr_t)xg;
        const unsigned int lds_addr = (unsigned int)(uintptr_t)(&sX[0][0]);

        u32x4 g0;
        g0[0] = 1u;                                     // count=1 (valid), user mode
        g0[1] = lds_addr;                               // D#.lds_addr
        g0[2] = (unsigned int)(ga & 0xFFFFFFFFull);     // global_addr[31:0]
        g0[3] = (unsigned int)((ga >> 32) & 0x1FFFFFFull) | (2u << 30); // addr[56:32] | type=2

        i32x8 g1;
        g1[0] = (2 << 16)        // data_size = 4 bytes
              | (1 << 20)        // pad_enable
              | (5 << 22)        // pad_interval: 64 DWORDs
              | (11 << 25);      // pad_amount: 12 DWORDs
        g1[1] = (FF & 0xFFFF) << 16;    // tensor_dim0 = 64 (bits 63:48)
        g1[2] = (128 & 0xFFFF) << 16;   // tensor_dim1 = 128 (bits 95:80)
        g1[3] = (FF & 0xFFFF) << 16;    // tile_dim0 = 64 (bits 127:112)
        g1[4] = 128;                    // tile_dim1 = 128; tile_dim2 = 0
        g1[5] = FF;                     // tensor_dim0_stride = 64 elements
        g1[6] = 0;
        g1[7] = 0;

        const i32x4 gz4 = {0, 0, 0, 0};
#if __clang_major__ >= 23
        const i32x8 gz8 = {0, 0, 0, 0, 0, 0, 0, 0};
        __builtin_amdgcn_tensor_load_to_lds(g0, g1, gz4, gz4, gz8, 0);
#else
        __builtin_amdgcn_tensor_load_to_lds(g0, g1, gz4, gz4, 0);
#endif
    }
#else
    for (int i = threadIdx.x; i < (128 * FF) / 4; i += 256) {
        float4 v4 = ((const float4*)xg)[i];
        int r = (4 * i) >> 6, c = (4 * i) & 63;
        sX[r][c] = v4.x; sX[r][c + 1] = v4.y; sX[r][c + 2] = v4.z; sX[r][c + 3] = v4.w;
    }
#endif

    // Stage W transposed: sWt[n][k] = W[k][n] so B fragments are contiguous.
    for (int i = threadIdx.x; i < (FF * FF) / 4; i += 256) {
        float4 v4 = ((const float4*)Wm)[i];
        int r = (4 * i) >> 6, c = (4 * i) & 63;
        sWt[c][r] = v4.x; sWt[c + 1][r] = v4.y; sWt[c + 2][r] = v4.z; sWt[c + 3][r] = v4.w;
    }

#if USE_TDM
    if (threadIdx.x == 0) __builtin_amdgcn_s_wait_tensorcnt(0);
#endif
    __syncthreads();

    const int wave  = threadIdx.x >> 5;   // 0..7: 16-row slab per wave
    const int lane  = threadIdx.x & 31;
    const int lm    = lane & 15;
    const int khalf = lane >> 4;          // 0: K-cols {0,1}; 1: K-cols {2,3}
    const int rbase = wave * 16;

    // Preload all 16 A fragments (row = rbase+lm, cols 4kk + 2*khalf .. +1)
    v2f a[16];
#pragma unroll
    for (int kk = 0; kk < 16; ++kk) {
        int kc = kk * 4 + khalf * 2;
        a[kk][0] = sX[rbase + lm][kc];
        a[kk][1] = sX[rbase + lm][kc + 1];
    }

#pragma unroll
    for (int nt = 0; nt < 4; ++nt) {
        const int n0 = nt * 16;
        v8f c = {0.f, 0.f, 0.f, 0.f, 0.f, 0.f, 0.f, 0.f};
#pragma unroll
        for (int kk = 0; kk < 16; ++kk) {
            int kr = kk * 4 + khalf * 2;    // B rows {kr, kr+1} = consecutive in sWt
            v2f b;
            b[0] = sWt[n0 + lm][kr];
            b[1] = sWt[n0 + lm][kr + 1];
            c = __builtin_amdgcn_wmma_f32_16x16x4_f32(
                    false, a[kk], false, b, (short)0, c, false, false);
        }
        float bb = bia[n0 + lm];
        // C layout: VGPR j -> M = j + 8*khalf, N = n0 + lm
        float* drow = dst + (((size_t)bt * NN) + row0 + rbase + 8 * khalf) * FF + n0 + lm;
#pragma unroll
        for (int j = 0; j < 8; ++j)
            drow[(size_t)j * FF] = c[j] + bb;
    }
}

// ---------------------------------------------------------------------------
// CSR build (deterministic: no atomic scatter; per-node ordered scan of edges)
// edge_index layout: ei[0..E-1] = src, ei[E..2E-1] = dst
// ---------------------------------------------------------------------------
__global__ __launch_bounds__(256)
void csr_count_kernel(const int* __restrict__ ei, int* __restrict__ counts)
{
    int node = blockIdx.x * 256 + threadIdx.x;
    if (node >= NN) return;
    const int* dstv = ei + EE;
    int c = 0;
#pragma unroll 4
    for (int e = 0; e < EE; ++e) c += (dstv[e] == node) ? 1 : 0;
    counts[node] = c;
}

__global__ void csr_scan_kernel(const int* __restrict__ counts, int* __restrict__ offsets)
{
    if (blockIdx.x == 0 && threadIdx.x == 0) {
        int run = 0;
        for (int i = 0; i < NN; ++i) { offsets[i] = run; run += counts[i]; }
        offsets[NN] = run;
    }
}

__global__ __launch_bounds__(256)
void csr_fill_kernel(const int* __restrict__ ei, const int* __restrict__ offsets,
                     int* __restrict__ srcs_sorted)
{
    int node = blockIdx.x * 256 + threadIdx.x;
    if (node >= NN) return;
    const int* srcv = ei;
    const int* dstv = ei + EE;
    int pos = offsets[node];
#pragma unroll 4
    for (int e = 0; e < EE; ++e)
        if (dstv[e] == node) srcs_sorted[pos++] = srcv[e];
}

// ---------------------------------------------------------------------------
// Kernel 3: per-node gather/reduce. One wave per (graph, node); lane owns 2
// features (float2). Q/V rows are L2-resident (75 MB << 192 MB L2).
//   out[bt,node,:] = skip(already in out) + sum_e lrelu(k_i + q_src)*v_src
// ---------------------------------------------------------------------------
__global__ __launch_bounds__(256)
void gather_kernel(const float* __restrict__ K, const float* __restrict__ Q,
                   const float* __restrict__ V, const int* __restrict__ offsets,
                   const int* __restrict__ srcs, float* __restrict__ out)
{
    const int wave = threadIdx.x >> 5;
    const int lane = threadIdx.x & 31;
    const int g    = blockIdx.x * 8 + wave;   // over BT*NN
    const int bt   = g >> 11;                  // / NN
    const int node = g & (NN - 1);

    const size_t gb = (size_t)bt * NN * FF;
    const float2* Krow = (const float2*)(K + gb + (size_t)node * FF);
    const float2* Qb   = (const float2*)(Q + gb);
    const float2* Vb   = (const float2*)(V + gb);

    const float2 ki = Krow[lane];
    float ax = 0.f, ay = 0.f;

    const int beg = offsets[node];
    const int end = offsets[node + 1];
    for (int e = beg; e < end; ++e) {
        const int s = srcs[e];
        if (e + 1 < end) {
            const int sn = srcs[e + 1];   // prefetch next edge's rows (L2 hit path)
            __builtin_prefetch((const void*)(Qb + (size_t)sn * (FF / 2) + lane), 0, 0);
            __builtin_prefetch((const void*)(Vb + (size_t)sn * (FF / 2) + lane), 0, 0);
        }
        const float2 qv = Qb[(size_t)s * (FF / 2) + lane];
        const float2 vv = Vb[(size_t)s * (FF / 2) + lane];
        float g0 = ki.x + qv.x;
        float g1 = ki.y + qv.y;
        g0 = (g0 > 0.f) ? g0 : 0.01f * g0;
        g1 = (g1 > 0.f) ? g1 : 0.01f * g1;
        ax = fmaf(g0, vv.x, ax);
        ay = fmaf(g1, vv.y, ay);
    }

    float2* orow = (float2*)(out + gb + (size_t)node * FF);
    const float2 sk = orow[lane];          // skip + bias written by proj kernel
    orow[lane] = make_float2(sk.x + ax, sk.y + ay);
}

// ---------------------------------------------------------------------------
extern "C" void kernel_launch(void* const* d_in, const int* in_sizes, int n_in,
                              void* d_out, int out_size, void* d_ws, size_t ws_size,
                              hipStream_t stream)
{
    const float* x  = (const float*)d_in[0];
    const int*   ei = (const int*)  d_in[1];
    const float* Wk = (const float*)d_in[2];
    const float* bk = (const float*)d_in[3];
    const float* Wq = (const float*)d_in[4];
    const float* bq = (const float*)d_in[5];
    const float* Wv = (const float*)d_in[6];
    const float* bv = (const float*)d_in[7];
    const float* Ws = (const float*)d_in[8];
    const float* bs = (const float*)d_in[9];
    float* out = (float*)d_out;

    // Workspace layout
    const size_t kqv = (size_t)BT * NN * FF;   // 6,291,456 floats each
    float* K = (float*)d_ws;
    float* Q = K + kqv;
    float* V = Q + kqv;
    int* counts  = (int*)(V + kqv);
    int* offsets = counts + NN;
    int* srcs    = offsets + (NN + 1);

    // 1) Projections (K, Q, V to ws; skip+bias directly into out)
    dim3 pg(BT * (NN / 128), 4);
    proj_wmma_kernel<<<pg, 256, 0, stream>>>(x, Wk, bk, Wq, bq, Wv, bv, Ws, bs,
                                             K, Q, V, out);

    // 2) CSR build (deterministic)
    csr_count_kernel<<<NN / 256, 256, 0, stream>>>(ei, counts);
    csr_scan_kernel<<<1, 32, 0, stream>>>(counts, offsets);
    csr_fill_kernel<<<NN / 256, 256, 0, stream>>>(ei, offsets, srcs);

    // 3) Edge gather + reduce + residual
    gather_kernel<<<(BT * NN) / 8, 256, 0, stream>>>(K, Q, V, offsets, srcs, out);
}